// VectorQuantizer_3813930959336
// MI455X (gfx1250) — compile-verified
//
#include <hip/hip_runtime.h>
#include <hip/hip_bf16.h>

typedef __attribute__((ext_vector_type(2))) float v2f;
typedef __attribute__((ext_vector_type(8))) float v8f;

#define KCB   4096
#define DM    512
#define NROWS 32768
#define BETA  0.05f
#define ROWS_PER_BLOCK 64

// Monotone map f32 -> u32 so unsigned-integer min == float min.
__device__ __forceinline__ unsigned orderf(float x) {
  unsigned u = __float_as_uint(x);
  return (u & 0x80000000u) ? ~u : (u | 0x80000000u);
}

// Global row (rt*16 + r15)  ->  LDS row (r15*4 + rt): keeps the 4 A-fragment
// bases of one lane within 8 KB so DS immediate offsets cover them.
__device__ __forceinline__ int lds_row(int r) { return ((r & 15) << 2) + (r >> 4); }

// ---------------------------------------------------------------------------
// Kernel 0: enorm[k] = sum_d emb[k][d]^2 ; also copy emb_weight to its output
// slot (the reference returns emb_weight as output #4).
// ---------------------------------------------------------------------------
__global__ void vq_enorm_kernel(const float* __restrict__ emb,
                                float* __restrict__ enorm,
                                float* __restrict__ emb_out) {
  __shared__ float red[64];
  const int k = blockIdx.x;
  const int t = threadIdx.x;
  const float* row = emb + (size_t)k * DM;
  float* orow = emb_out + (size_t)k * DM;
  float s = 0.f;
#pragma unroll
  for (int i = 0; i < 2; ++i) {
    float4 v = *(const float4*)(row + t * 4 + i * 256);
    *(float4*)(orow + t * 4 + i * 256) = v;
    s += v.x * v.x + v.y * v.y + v.z * v.z + v.w * v.w;
  }
  red[t] = s;
  __syncthreads();
#pragma unroll
  for (int off = 32; off > 0; off >>= 1) {
    if (t < off) red[t] += red[t + off];
    __syncthreads();
  }
  if (t == 0) enorm[k] = red[0];
}

// ---------------------------------------------------------------------------
// Kernel 1: fused distance-GEMM (f32 WMMA) + argmin + gather + loss + counts.
// One block = 8 waves owns 64 latent rows (4 row-tiles of 16). Each wave
// computes 16 codebook columns per chunk; one B fragment feeds 4 WMMAs
// (one per row-tile), so codebook/L2 traffic is amortized 4x.
// ---------------------------------------------------------------------------
__global__ void __launch_bounds__(256)
vq_argmin_kernel(const float* __restrict__ latents,
                 const float* __restrict__ emb,
                 const float* __restrict__ enorm_g,
                 float* __restrict__ q_out,
                 float* __restrict__ inds_out,
                 float* __restrict__ counts,
                 float* __restrict__ loss_accum) {
  __shared__ float Xl[ROWS_PER_BLOCK * DM];            // 128 KB X tile (interleaved rows)
  __shared__ float enl[KCB];                           // 16 KB ||e_k||^2
  __shared__ unsigned long long minkey[ROWS_PER_BLOCK];
  __shared__ unsigned rowind[ROWS_PER_BLOCK];
  __shared__ float lsum;

  const int tid = threadIdx.x;
  const int row0 = blockIdx.x * ROWS_PER_BLOCK;

  // Stage X tile: thread t -> global row t/4 (stored permuted), 128 floats.
  {
    const int r = tid >> 2, c = (tid & 3) * 128;
    const float* xr = latents + (size_t)(row0 + r) * DM + c;
    float* xd = Xl + lds_row(r) * DM + c;
#pragma unroll
    for (int i = 0; i < 32; ++i)
      *(float4*)(xd + i * 4) = *(const float4*)(xr + i * 4);
  }
  // Stage enorm (4096 floats / 256 threads = 16 each).
  {
#pragma unroll
    for (int i = 0; i < 4; ++i) {
      const int o = tid * 16 + i * 4;
      *(float4*)(enl + o) = *(const float4*)(enorm_g + o);
    }
  }
  if (tid < ROWS_PER_BLOCK) minkey[tid] = 0xFFFFFFFFFFFFFFFFull;
  if (tid == 0) lsum = 0.f;
  __syncthreads();

  const int lane = tid & 31;
  const int wave = tid >> 5;
  const int l15 = lane & 15;
  const int half = lane >> 4;        // 0: rows +0..7 / k{0,1}; 1: rows +8..15 / k{2,3}
  const int off = half << 1;

  // A fragment base: lane row r15=l15 -> LDS rows l15*4 + rt, rt stride = DM floats.
  const float* xl = Xl + (l15 << 2) * DM + off;

  for (int k0 = wave * 16; k0 < KCB; k0 += 128) {
    const int col = k0 + l15;
    const float* ep = emb + (size_t)col * DM + off;   // B fragment source
    v8f acc0 = {0.f, 0.f, 0.f, 0.f, 0.f, 0.f, 0.f, 0.f};
    v8f acc1 = acc0, acc2 = acc0, acc3 = acc0;
#pragma unroll 4
    for (int d = 0; d < DM; d += 4) {
      v2f b;  b.x  = ep[d];              b.y  = ep[d + 1];
      v2f a0; a0.x = xl[d];              a0.y = xl[d + 1];
      v2f a1; a1.x = xl[1 * DM + d];     a1.y = xl[1 * DM + d + 1];
      v2f a2; a2.x = xl[2 * DM + d];     a2.y = xl[2 * DM + d + 1];
      v2f a3; a3.x = xl[3 * DM + d];     a3.y = xl[3 * DM + d + 1];
      acc0 = __builtin_amdgcn_wmma_f32_16x16x4_f32(false, a0, false, b, (short)0, acc0, false, false);
      acc1 = __builtin_amdgcn_wmma_f32_16x16x4_f32(false, a1, false, b, (short)0, acc1, false, false);
      acc2 = __builtin_amdgcn_wmma_f32_16x16x4_f32(false, a2, false, b, (short)0, acc2, false, false);
      acc3 = __builtin_amdgcn_wmma_f32_16x16x4_f32(false, a3, false, b, (short)0, acc3, false, false);
    }
    const float en = enl[col];
#pragma unroll
    for (int rt = 0; rt < 4; ++rt) {
      const v8f& acc = (rt == 0) ? acc0 : (rt == 1) ? acc1 : (rt == 2) ? acc2 : acc3;
      float sc[8]; unsigned ix[8];
#pragma unroll
      for (int j = 0; j < 8; ++j) { sc[j] = en - 2.f * acc[j]; ix[j] = (unsigned)col; }
      // Per-row (value,index) min across the 16 lanes of each half-wave.
#pragma unroll
      for (int m = 1; m < 16; m <<= 1) {
#pragma unroll
        for (int j = 0; j < 8; ++j) {
          float ov = __shfl_xor(sc[j], m, 32);
          unsigned oi = (unsigned)__shfl_xor((int)ix[j], m, 32);
          if (ov < sc[j] || (ov == sc[j] && oi < ix[j])) { sc[j] = ov; ix[j] = oi; }
        }
      }
      if (l15 == 0) {
        const int rbase = rt * 16 + half * 8;   // global row within block
#pragma unroll
        for (int j = 0; j < 8; ++j) {
          unsigned long long key =
              ((unsigned long long)orderf(sc[j]) << 32) | (unsigned long long)ix[j];
          atomicMin(&minkey[rbase + j], key);   // ds_min_u64, low bits break ties
        }
      }
    }
  }
  __syncthreads();

  if (tid < ROWS_PER_BLOCK) {
    const unsigned idx = (unsigned)(minkey[tid] & 0xFFFFFFFFull);
    rowind[tid] = idx;
    inds_out[row0 + tid] = (float)idx;
    atomicAdd(&counts[idx], 1.0f);
  }
  __syncthreads();

  // Gather winning codes, write quantized_st, accumulate exact sum (q-x)^2.
  {
    const int r = tid >> 2, c = (tid & 3) * 128;
    const float* ep = emb + (size_t)rowind[r] * DM + c;
    const float* xp = Xl + lds_row(r) * DM + c;
    float* qp = q_out + (size_t)(row0 + r) * DM + c;
    float s = 0.f;
#pragma unroll
    for (int i = 0; i < 32; ++i) {
      float4 q = *(const float4*)(ep + i * 4);
      float4 x = *(const float4*)(xp + i * 4);
      *(float4*)(qp + i * 4) = q;
      float dx = q.x - x.x, dy = q.y - x.y, dz = q.z - x.z, dw = q.w - x.w;
      s += dx * dx + dy * dy + dz * dz + dw * dw;
    }
    atomicAdd(&lsum, s);               // ds_add_f32: block-local reduction
  }
  __syncthreads();
  if (tid == 0) atomicAdd(loss_accum, lsum);   // 1 global atomic per block
}

// ---------------------------------------------------------------------------
// Kernel 2: perplexity from counts, final loss scale.
// ---------------------------------------------------------------------------
__global__ void vq_finalize_kernel(const float* __restrict__ counts,
                                   const float* __restrict__ loss_accum,
                                   float* __restrict__ loss_out,
                                   float* __restrict__ perp_out) {
  __shared__ float red[256];
  const int t = threadIdx.x;
  float s = 0.f;
  for (int k = t; k < KCB; k += 256) {
    float p = counts[k] * (1.0f / (float)NROWS);
    s += p * __logf(p + 1e-10f);
  }
  red[t] = s;
  __syncthreads();
#pragma unroll
  for (int off = 128; off > 0; off >>= 1) {
    if (t < off) red[t] += red[t + off];
    __syncthreads();
  }
  if (t == 0) {
    perp_out[0] = __expf(-red[0]);
    loss_out[0] =
        loss_accum[0] * (1.0f + BETA) / (float)((long long)NROWS * (long long)DM);
  }
}

// ---------------------------------------------------------------------------
extern "C" void kernel_launch(void* const* d_in, const int* in_sizes, int n_in,
                              void* d_out, int out_size, void* d_ws, size_t ws_size,
                              hipStream_t stream) {
  (void)in_sizes; (void)n_in; (void)out_size; (void)ws_size;
  const float* latents = (const float*)d_in[0];   // [32,1024,512] f32
  const float* emb     = (const float*)d_in[1];   // [4096,512]    f32

  float* out      = (float*)d_out;
  float* q_out    = out;                                   // 32768*512
  float* loss_out = out + (size_t)NROWS * DM;              // 1
  float* inds_out = loss_out + 1;                          // 32768
  float* emb_out  = inds_out + NROWS;                      // 4096*512
  float* perp_out = emb_out + (size_t)KCB * DM;            // 1

  float* enorm      = (float*)d_ws;        // KCB floats
  float* counts     = enorm + KCB;         // KCB floats
  float* loss_accum = counts + KCB;        // 1 float

  hipMemsetAsync(counts, 0, (KCB + 1) * sizeof(float), stream);
  vq_enorm_kernel<<<KCB, 64, 0, stream>>>(emb, enorm, emb_out);
  vq_argmin_kernel<<<NROWS / ROWS_PER_BLOCK, 256, 0, stream>>>(
      latents, emb, enorm, q_out, inds_out, counts, loss_accum);
  vq_finalize_kernel<<<1, 256, 0, stream>>>(counts, loss_accum, loss_out, perp_out);
}